// GCNConvNet_1434519076955
// MI455X (gfx1250) — compile-verified
//
#include <hip/hip_runtime.h>

typedef __attribute__((ext_vector_type(2))) float v2f;
typedef __attribute__((ext_vector_type(8))) float v8f;
typedef __attribute__((ext_vector_type(4))) unsigned int tdm_u32x4;
typedef __attribute__((ext_vector_type(8))) int tdm_i32x8;
typedef __attribute__((ext_vector_type(4))) int tdm_i32x4;

#define NNODES 50000
#define LRELU  0.01f
#define EPSBN  1e-5f

// ---------------- degree / dinv ----------------
__global__ void k_init_deg(float* deg, int n) {
  int i = blockIdx.x * blockDim.x + threadIdx.x;
  if (i < n) deg[i] = 1.0f;  // self-loop contributes 1
}
__global__ void k_edge_deg(const int* __restrict__ dst, float* __restrict__ deg, int e) {
  int i = blockIdx.x * blockDim.x + threadIdx.x;
  if (i < e) atomicAdd(&deg[dst[i]], 1.0f);
}
__global__ void k_rsqrt_inplace(float* d, int n) {
  int i = blockIdx.x * blockDim.x + threadIdx.x;
  if (i < n) d[i] = rsqrtf(d[i]);
}
__global__ void k_zero(float* p, int n) {
  int i = blockIdx.x * blockDim.x + threadIdx.x;
  if (i < n) p[i] = 0.0f;
}

// ---------------- TDM: 2D fp32 tile -> LDS, optional per-row padding ----------------
// D# per CDNA5 ISA ch.8: group0 = {count/type/addr}, group1 = {data_size, dims,
// tile dims, dim0 stride, pad}. pad_interval code c => pad every (2<<c) dwords.
// This toolchain uses the 6-arg builtin: (u32x4, i32x8, i32x4, i32x4, i32x8, cpol).
__device__ __forceinline__ void tdm_load_tile_f32(
    void* lds_dst, const float* gsrc, unsigned k_elems, unsigned rows,
    unsigned row_stride_elems, unsigned pad_interval_code, unsigned pad_dwords) {
#if __has_builtin(__builtin_amdgcn_tensor_load_to_lds)
  unsigned long long ga = (unsigned long long)(uintptr_t)gsrc;
  unsigned lds_addr = (unsigned)(uintptr_t)lds_dst;  // low 32 bits = LDS byte offset
  tdm_u32x4 g0;
  g0.x = 1u;                                          // count=1 (valid), user mode
  g0.y = lds_addr;                                    // bits 63:32  lds_addr
  g0.z = (unsigned)ga;                                // bits 95:64  global_addr[31:0]
  g0.w = (unsigned)((ga >> 32) & 0x01ffffffu) | (2u << 30);  // addr[56:32] | type=2
  unsigned pad_en = pad_dwords ? 1u : 0u;
  tdm_i32x8 g1;
  g1[0] = (int)((2u << 16) |                          // data_size = 4 bytes
                (pad_en << 20) |
                ((pad_interval_code & 7u) << 22) |
                (((pad_dwords ? pad_dwords - 1u : 0u) & 0x7fu) << 25));
  g1[1] = (int)((k_elems & 0xffffu) << 16);           // tensor_dim0[15:0]
  g1[2] = (int)(((k_elems >> 16) & 0xffffu) |         // tensor_dim0[31:16]
                ((rows & 0xffffu) << 16));            // tensor_dim1[15:0]
  g1[3] = (int)(((rows >> 16) & 0xffffu) |            // tensor_dim1[31:16]
                ((k_elems & 0xffffu) << 16));         // tile_dim0
  g1[4] = (int)(rows & 0xffffu);                      // tile_dim1 (tile_dim2 = 0)
  g1[5] = (int)row_stride_elems;                      // tensor_dim0_stride[31:0]
  g1[6] = 0;                                          // stride hi | dim1_stride lo
  g1[7] = 0;
  tdm_i32x4 gz4 = {0, 0, 0, 0};
  tdm_i32x8 gz8 = {0, 0, 0, 0, 0, 0, 0, 0};
  __builtin_amdgcn_tensor_load_to_lds(g0, g1, gz4, gz4, gz8, 0);
#endif
}

// ---------------- fp32 WMMA GEMM: H[M,O] = A[M,K] @ W[O,K]^T ----------------
// K is a compile-time constant (128 or 256): fully unrolled WMMA chain, no
// remainder loop. Per 256-thread block: 8 waves, one 16x16 tile each
// (8 M-tiles x 1 O-tile). TDM stages the block's 16xK W slab (shared) and each
// wave's 16xK A slab into LDS with a +4-dword row pad (bank-conflict-free b64
// reads); WMMA consumes from LDS via ds_load_b64.
template <int K>
__global__ __launch_bounds__(256)
void k_gemm_wmma(const float* __restrict__ A, const float* __restrict__ W,
                 float* __restrict__ H, int M, int O) {
  int mt = blockIdx.y * 8 + (threadIdx.x >> 5);   // wave -> M tile
  if (mt * 16 >= M) return;                        // wave-uniform guard
  int ot   = blockIdx.x;                           // O tile
  int w    = threadIdx.x >> 5;
  int lane = threadIdx.x & 31;
  int r    = lane & 15;
  int kh   = (lane >> 4) << 1;                     // K offset 0 or 2

  extern __shared__ float smem[];
  constexpr int Ks = K + 4;                        // padded LDS row stride (dwords)
  float* ldsB = smem;                              // 16 * Ks
  float* ldsA = smem + 16 * Ks;                    // 8 waves * 16 * Ks

#if __has_builtin(__builtin_amdgcn_tensor_load_to_lds)
  constexpr unsigned padcode = (K == 256) ? 7u : 6u;  // pad every K dwords
  tdm_load_tile_f32(ldsA + w * 16 * Ks, A + (size_t)mt * 16 * K,
                    (unsigned)K, 16u, (unsigned)K, padcode, 4u);
  if (w == 0)
    tdm_load_tile_f32(ldsB, W + (size_t)ot * 16 * K,
                      (unsigned)K, 16u, (unsigned)K, padcode, 4u);
  __builtin_amdgcn_s_wait_tensorcnt(0);
  __syncthreads();                                 // publish W slab to all waves
#else
  // Fallback: cooperative plain copies into the same padded layout.
  constexpr int kshift = (K == 256) ? 8 : 7;
  for (int idx = threadIdx.x; idx < 16 * K; idx += 256) {
    int rr = idx >> kshift, kk = idx & (K - 1);
    ldsB[rr * Ks + kk] = W[(size_t)(ot * 16 + rr) * K + kk];
  }
  for (int idx = lane; idx < 16 * K; idx += 32) {
    int rr = idx >> kshift, kk = idx & (K - 1);
    ldsA[w * 16 * Ks + rr * Ks + kk] = A[(size_t)(mt * 16 + rr) * K + kk];
  }
  __syncthreads();
#endif

  const float* aRow = ldsA + w * 16 * Ks + r * Ks + kh;
  const float* bRow = ldsB + r * Ks + kh;

  v8f acc = {};
#if __has_builtin(__builtin_amdgcn_wmma_f32_16x16x4_f32)
  #pragma unroll
  for (int k = 0; k < K; k += 4) {
    v2f a = *(const v2f*)(aRow + k);
    v2f b = *(const v2f*)(bRow + k);
    acc = __builtin_amdgcn_wmma_f32_16x16x4_f32(
        /*neg_a=*/false, a, /*neg_b=*/false, b,
        /*c_mod=*/(short)0, acc, /*reuse_a=*/false, /*reuse_b=*/false);
  }
#else
  int colf = ot * 16 + (lane & 15);
  int rowf = mt * 16 + ((lane >> 4) << 3);
  #pragma unroll
  for (int i = 0; i < 8; ++i) {
    float s = 0.f;
    for (int k = 0; k < K; ++k)
      s = fmaf(A[(size_t)(rowf + i) * K + k], W[(size_t)colf * K + k], s);
    acc[i] = s;
  }
#endif

  int row0 = mt * 16 + ((lane >> 4) << 3);
  float* out = H + (size_t)ot * 16 + (lane & 15);
  #pragma unroll
  for (int i = 0; i < 8; ++i) out[(size_t)(row0 + i) * O] = acc[i];
}

// ---------------- AGG init: bias + self-loop term ----------------
__global__ __launch_bounds__(256)
void k_init_agg(const float* __restrict__ H, const float* __restrict__ b,
                const float* __restrict__ dinv, float* __restrict__ AGG,
                int total4, int cshift, int cmask) {
  int t = blockIdx.x * blockDim.x + threadIdx.x;
  if (t >= total4) return;
  int p   = t << 2;
  int row = p >> cshift;
  int col = p & cmask;
  float di = dinv[row];
  float wsl = di * di;                      // self-loop enorm
  float4 h4 = *(const float4*)(H + p);
  float4 b4 = *(const float4*)(b + col);
  float4 o;
  o.x = fmaf(wsl, h4.x, b4.x);
  o.y = fmaf(wsl, h4.y, b4.y);
  o.z = fmaf(wsl, h4.z, b4.z);
  o.w = fmaf(wsl, h4.w, b4.w);
  *(float4*)(AGG + p) = o;
}

// ---------------- edge scatter: AGG[dst] += H[src] * dinv[src]*dinv[dst] ----------------
__global__ __launch_bounds__(256)
void k_scatter(const float* __restrict__ H, const int* __restrict__ src,
               const int* __restrict__ dst, const float* __restrict__ dinv,
               float* __restrict__ AGG, int E, int cpeShift, int C) {
  int local = threadIdx.x;
  int e = blockIdx.x * (256 >> cpeShift) + (local >> cpeShift);
  if (e >= E) return;
  int chunk = local & ((1 << cpeShift) - 1);
  int s = src[e], d = dst[e];
  float wgt = dinv[s] * dinv[d];
  float4 h4 = *(const float4*)(H + (size_t)s * C + (chunk << 2));
  float* o = AGG + (size_t)d * C + (chunk << 2);
  atomicAdd(o + 0, h4.x * wgt);
  atomicAdd(o + 1, h4.y * wgt);
  atomicAdd(o + 2, h4.z * wgt);
  atomicAdd(o + 3, h4.w * wgt);
}

// ---------------- BatchNorm column stats ----------------
__global__ void k_colstats(const float* __restrict__ X, int rows, int C,
                           float* __restrict__ sum, float* __restrict__ sumsq) {
  int c = threadIdx.x;  // blockDim.x == C
  float s = 0.f, q = 0.f;
  for (int r = blockIdx.x; r < rows; r += gridDim.x) {
    __builtin_prefetch(X + (size_t)(r + gridDim.x) * C + c, 0, 0);
    float v = X[(size_t)r * C + c];
    s += v;
    q = fmaf(v, v, q);
  }
  atomicAdd(&sum[c], s);
  atomicAdd(&sumsq[c], q);
}

__global__ void k_finstats(const float* __restrict__ sum, const float* __restrict__ sumsq,
                           const float* __restrict__ g, const float* __restrict__ be,
                           float* __restrict__ scale, float* __restrict__ shift,
                           float invn, int C) {
  int c = threadIdx.x;
  if (c >= C) return;
  float mu  = sum[c] * invn;
  float var = fmaf(-mu, mu, sumsq[c] * invn);   // biased variance
  float sc  = g[c] * rsqrtf(var + EPSBN);
  scale[c] = sc;
  shift[c] = fmaf(-mu, sc, be[c]);
}

// ---------------- normalize + optional leaky relu ----------------
__global__ __launch_bounds__(256)
void k_normact(const float* __restrict__ AGG, const float* __restrict__ scale,
               const float* __restrict__ shift, float* __restrict__ OUT,
               int total4, int cmask, int leaky) {
  int t = blockIdx.x * blockDim.x + threadIdx.x;
  if (t >= total4) return;
  int p   = t << 2;
  int col = p & cmask;
  float4 v = *(const float4*)(AGG + p);
  float4 o;
  o.x = fmaf(v.x, scale[col + 0], shift[col + 0]);
  o.y = fmaf(v.y, scale[col + 1], shift[col + 1]);
  o.z = fmaf(v.z, scale[col + 2], shift[col + 2]);
  o.w = fmaf(v.w, scale[col + 3], shift[col + 3]);
  if (leaky) {
    o.x = o.x > 0.f ? o.x : o.x * LRELU;
    o.y = o.y > 0.f ? o.y : o.y * LRELU;
    o.z = o.z > 0.f ? o.z : o.z * LRELU;
    o.w = o.w > 0.f ? o.w : o.w * LRELU;
  }
  *(float4*)(OUT + p) = o;
}

extern "C" void kernel_launch(void* const* d_in, const int* in_sizes, int n_in,
                              void* d_out, int out_size, void* d_ws, size_t ws_size,
                              hipStream_t stream) {
  const float* x  = (const float*)d_in[0];
  const int*   ei = (const int*)d_in[1];
  const int E = in_sizes[1] / 2;
  const int* src = ei;        // edge_index[0]
  const int* dst = ei + E;    // edge_index[1]

  const float* W0 = (const float*)d_in[2];
  const float* b0 = (const float*)d_in[3];
  const float* g0 = (const float*)d_in[4];
  const float* be0= (const float*)d_in[5];
  const float* W1 = (const float*)d_in[6];
  const float* b1 = (const float*)d_in[7];
  const float* g1 = (const float*)d_in[8];
  const float* be1= (const float*)d_in[9];
  const float* W2 = (const float*)d_in[10];
  const float* b2 = (const float*)d_in[11];
  const float* g2 = (const float*)d_in[12];
  const float* be2= (const float*)d_in[13];

  float* ws   = (float*)d_ws;
  float* H    = ws;                               // N*256
  float* AGG  = H   + (size_t)NNODES * 256;       // N*256
  float* X1   = AGG + (size_t)NNODES * 256;       // N*256 (activations)
  float* dinv = X1  + (size_t)NNODES * 256;       // N
  float* sum  = dinv + NNODES;                    // 256
  float* sumsq= sum + 256;                        // 256
  float* scale= sumsq + 256;                      // 256
  float* shift= scale + 256;                      // 256

  // degrees -> dinv
  k_init_deg<<<(NNODES + 255) / 256, 256, 0, stream>>>(dinv, NNODES);
  k_edge_deg<<<(E + 255) / 256, 256, 0, stream>>>(dst, dinv, E);
  k_rsqrt_inplace<<<(NNODES + 255) / 256, 256, 0, stream>>>(dinv, NNODES);

  auto layer = [&](const float* X, int K, const float* W, const float* b,
                   const float* g, const float* be, int C, float* OUT, int leaky) {
    const int mtiles = NNODES / 16;  // 3125 (exact)
    dim3 ggrid(C / 16, (mtiles + 7) / 8);
    size_t smem = (size_t)(8 * 16 + 16) * (K + 4) * sizeof(float); // A slabs + W slab
    if (K == 128)
      k_gemm_wmma<128><<<ggrid, 256, smem, stream>>>(X, W, H, NNODES, C);
    else
      k_gemm_wmma<256><<<ggrid, 256, smem, stream>>>(X, W, H, NNODES, C);

    const int total4 = NNODES * C / 4;
    const int cshift = (C == 256) ? 8 : 7;
    const int cmask  = C - 1;
    k_init_agg<<<(total4 + 255) / 256, 256, 0, stream>>>(H, b, dinv, AGG, total4, cshift, cmask);

    const int cpeShift = (C == 256) ? 6 : 5;     // float4 chunks per edge: 64 or 32
    const int epb = 256 >> cpeShift;             // edges per 256-thread block
    k_scatter<<<(E + epb - 1) / epb, 256, 0, stream>>>(H, src, dst, dinv, AGG, E, cpeShift, C);

    k_zero<<<2, 256, 0, stream>>>(sum, 512);     // zero sum[256] + sumsq[256]
    k_colstats<<<512, C, 0, stream>>>(AGG, NNODES, C, sum, sumsq);
    k_finstats<<<1, C, 0, stream>>>(sum, sumsq, g, be, scale, shift, 1.0f / NNODES, C);
    k_normact<<<(total4 + 255) / 256, 256, 0, stream>>>(AGG, scale, shift, OUT, total4, cmask, leaky);
  };

  layer(x,  128, W0, b0, g0, be0, 256, X1, 1);
  layer(X1, 256, W1, b1, g1, be1, 256, X1, 1);
  layer(X1, 256, W2, b2, g2, be2, 128, (float*)d_out, 0);
}